// SpatialGraphConv_6648609374249
// MI455X (gfx1250) — compile-verified
//
#include <hip/hip_runtime.h>
#include <hip/hip_bf16.h>

// SpatialGraphConv fused: xg = x*adj (per-(b,c,t) 25-vec x 25x25), z = W*xg,
// BN(train) + ReLU. Conv bias cancels under BN (constant shift removed by the
// mean subtraction), so both GEMM passes compute z only.
//
// Pass structure (all on `stream`, graph-capture safe):
//   zero_accum      : zero 256-float stat accumulators in d_ws
//   gcn_gemm<1>     : fused agg + WMMA GEMM, accumulate per-channel sum/sumsq
//   finalize_stats  : scale = gamma*rsqrt(var+eps), shift = beta - scale*mean
//   gcn_gemm<0>     : recompute GEMM, apply relu(z*scale+shift), store out

typedef float v2f __attribute__((ext_vector_type(2)));
typedef float v8f __attribute__((ext_vector_type(8)));

#define BB_   64
#define CIN_  64
#define COUT_ 128
#define TT_   300
#define VV_   25
#define TC_   8                 // t values per block
#define NS_   (TC_ * VV_)       // 200 valid columns per block
#define NP_   208               // padded to 13 tiles of 16
#define R2_   208               // float2 per channel-pair row (== NP_; 208 % 32 == 16 -> conflict-free reads)
#define NTIL_ 13
#define NCHK_ ((TT_ + TC_ - 1) / TC_)   // 38 t-chunks
#define NTOT_ (BB_ * TT_ * VV_)         // 480000 (BN reduction size)
#define EPS_  1e-5f

#define WMMA_F32(A, Bf, C) \
  __builtin_amdgcn_wmma_f32_16x16x4_f32(false, (A), false, (Bf), (short)0, (C), false, false)

// STATS=1: accumulate per-channel sum/sumsq of z into accum[0..255]
// STATS=0: apply per-channel scale/shift (+ReLU), store to out
template <int STATS>
__global__ __launch_bounds__(256) void gcn_gemm(
    const float* __restrict__ x, const float* __restrict__ adj,
    const float* __restrict__ W, const float* __restrict__ coef,
    float* __restrict__ accum, float* __restrict__ out) {
  __shared__ float s_adj[VV_ * VV_];
  // xg stored channel-pair interleaved: element (c, n) at [((c>>1)*R2_ + n)*2 + (c&1)]
  // -> B fragment (c=4k+2*hl .. +1, n) is ONE ds_load_b64 at pair-row 2k+hl.
  __shared__ float s_xg[(CIN_ / 2) * R2_ * 2];   // 53.2 KB

  const int tid = threadIdx.x;
  const int bb = blockIdx.x / NCHK_;
  const int tc = blockIdx.x % NCHK_;
  const int t0 = tc * TC_;
  const int nvalid = min(TC_, TT_ - t0) * VV_;   // 200 or 100 (tail chunk)

  // stage adjacency (broadcast reads later)
  for (int i = tid; i < VV_ * VV_; i += 256) s_adj[i] = adj[i];
  // zero pad columns [200,208) of every channel-pair row (both interleaved slots)
  for (int i = tid; i < (CIN_ / 2) * (NP_ - NS_); i += 256) {
    int p = i >> 3;
    int idx = (p * R2_ + NS_ + (i & 7)) * 2;
    s_xg[idx] = 0.0f;
    s_xg[idx + 1] = 0.0f;
  }
  __syncthreads();

  // ---- phase 1: graph aggregation into LDS (row-local: xg row = x row * adj)
  // Accumulate 25 outputs in registers, sweeping s_adj rows consecutively so
  // the unrolled reads fuse into ds_load_b128.
  for (int p = 0; p < 2; ++p) {
    int row = tid + p * 256;          // 512 (c, t_local) rows
    int c = row >> 3, tl = row & 7;
    int t = t0 + tl;
    const float* xp = x + (((size_t)bb * CIN_ + c) * TT_ + t) * VV_;
    float xr[VV_];
#pragma unroll
    for (int v = 0; v < VV_; ++v) xr[v] = (t < TT_) ? xp[v] : 0.0f;
    float acc[VV_];
#pragma unroll
    for (int w = 0; w < VV_; ++w) acc[w] = 0.0f;
#pragma unroll
    for (int v = 0; v < VV_; ++v) {
      float a = xr[v];
#pragma unroll
      for (int w = 0; w < VV_; ++w) acc[w] = fmaf(a, s_adj[v * VV_ + w], acc[w]);
    }
    int base = ((c >> 1) * R2_ + tl * VV_) * 2 + (c & 1);
#pragma unroll
    for (int w = 0; w < VV_; ++w) s_xg[base + 2 * w] = acc[w];
  }
  __syncthreads();

  // ---- phase 2: z = W(128x64) * xg(64x208) via V_WMMA_F32_16X16X4_F32
  const int wave = tid >> 5;   // 8 waves = 8 M-tiles of 16
  const int lane = tid & 31;
  const int hl = lane >> 4;    // half-wave: K split for A/B frags, M split for C
  const int l15 = lane & 15;

  // A fragments: 16x4 per K-step; VGPR0 = K{0,2}, VGPR1 = K{1,3} (ISA 7.12.2)
  v2f afrag[16];
  {
    const float* wp = W + ((size_t)(wave * 16 + l15)) * CIN_ + hl * 2;
#pragma unroll
    for (int k = 0; k < 16; ++k) afrag[k] = *(const v2f*)(wp + 4 * k);
  }

  float csc[8], csh[8];
  if (!STATS) {
#pragma unroll
    for (int r = 0; r < 8; ++r) {
      int o = wave * 16 + hl * 8 + r;
      csc[r] = coef[o];
      csh[r] = coef[COUT_ + o];
    }
  }

  float vsum[8] = {0, 0, 0, 0, 0, 0, 0, 0};
  float vsq[8] = {0, 0, 0, 0, 0, 0, 0, 0};

  auto epilogue = [&](const v8f& c, int nb) {
    const bool valid = nb < nvalid;
    if (STATS) {
#pragma unroll
      for (int r = 0; r < 8; ++r) {
        float z = valid ? c[r] : 0.0f;
        vsum[r] += z;
        vsq[r] = fmaf(z, z, vsq[r]);
      }
    } else {
      if (valid) {
        int t = nb / VV_;
        int v = nb - t * VV_;
#pragma unroll
        for (int r = 0; r < 8; ++r) {
          int o = wave * 16 + hl * 8 + r;
          float val = fmaf(c[r], csc[r], csh[r]);
          out[(((size_t)bb * COUT_ + o) * TT_ + (t0 + t)) * VV_ + v] =
              val > 0.0f ? val : 0.0f;
        }
      }
    }
  };

  // Exactly two concurrent accumulator chains: enough to hide WMMA latency
  // (plus 8 waves/block) without spilling past 256 VGPRs into MSB-switch land.
#pragma unroll 1
  for (int nt = 0; nt < NTIL_ - 1; nt += 2) {
    const int nb0 = nt * 16 + l15;
    const int nb1 = nb0 + 16;
    v8f c0 = {0.f, 0.f, 0.f, 0.f, 0.f, 0.f, 0.f, 0.f};
    v8f c1 = {0.f, 0.f, 0.f, 0.f, 0.f, 0.f, 0.f, 0.f};
#pragma unroll
    for (int k = 0; k < 16; ++k) {
      const int pr = (2 * k + hl) * R2_;
      v2f b0 = *(const v2f*)&s_xg[(pr + nb0) * 2];   // single ds_load_b64
      v2f b1 = *(const v2f*)&s_xg[(pr + nb1) * 2];
      c0 = WMMA_F32(afrag[k], b0, c0);
      c1 = WMMA_F32(afrag[k], b1, c1);
    }
    epilogue(c0, nb0);
    epilogue(c1, nb1);
  }
  {  // tail tile 12
    const int nb = (NTIL_ - 1) * 16 + l15;
    v8f c = {0.f, 0.f, 0.f, 0.f, 0.f, 0.f, 0.f, 0.f};
#pragma unroll
    for (int k = 0; k < 16; ++k) {
      const int pr = (2 * k + hl) * R2_;
      v2f bf = *(const v2f*)&s_xg[(pr + nb) * 2];
      c = WMMA_F32(afrag[k], bf, c);
    }
    epilogue(c, nb);
  }

  if (STATS) {
#pragma unroll
    for (int r = 0; r < 8; ++r) {
      float s = vsum[r], q = vsq[r];
#pragma unroll
      for (int m = 8; m >= 1; m >>= 1) {   // reduce within each 16-lane half
        s += __shfl_xor(s, m, 16);
        q += __shfl_xor(q, m, 16);
      }
      if (l15 == 0) {
        int o = wave * 16 + hl * 8 + r;
        atomicAdd(&accum[o], s);            // global_atomic_add_f32
        atomicAdd(&accum[COUT_ + o], q);
      }
    }
  }
}

__global__ void zero_accum(float* __restrict__ accum) {
  accum[threadIdx.x] = 0.0f;
}

__global__ void finalize_stats(const float* __restrict__ accum,
                               const float* __restrict__ gamma,
                               const float* __restrict__ beta,
                               float* __restrict__ coef) {
  int o = threadIdx.x;
  if (o < COUT_) {
    const float invN = 1.0f / (float)NTOT_;
    float mean = accum[o] * invN;
    float var = fmaf(-mean, mean, accum[COUT_ + o] * invN);
    float inv = rsqrtf(var + EPS_);
    float sc = gamma[o] * inv;
    coef[o] = sc;
    coef[COUT_ + o] = fmaf(-sc, mean, beta[o]);   // conv bias cancels under BN
  }
}

extern "C" void kernel_launch(void* const* d_in, const int* in_sizes, int n_in,
                              void* d_out, int out_size, void* d_ws, size_t ws_size,
                              hipStream_t stream) {
  (void)in_sizes; (void)n_in; (void)out_size; (void)ws_size;
  const float* x = (const float*)d_in[0];
  const float* adj = (const float*)d_in[1];
  const float* W = (const float*)d_in[2];
  // d_in[3] = conv bias: mathematically cancelled by BatchNorm mean removal
  const float* gamma = (const float*)d_in[4];
  const float* beta = (const float*)d_in[5];
  float* out = (float*)d_out;

  float* accum = (float*)d_ws;        // 256 floats: sum[128], sumsq[128]
  float* coef = accum + 2 * COUT_;    // 256 floats: scale[128], shift[128]

  dim3 grid(BB_ * NCHK_);             // 64 batches x 38 t-chunks = 2432 blocks
  zero_accum<<<1, 2 * COUT_, 0, stream>>>(accum);
  gcn_gemm<1><<<grid, 256, 0, stream>>>(x, adj, W, coef, accum, nullptr);
  finalize_stats<<<1, COUT_, 0, stream>>>(accum, gamma, beta, coef);
  gcn_gemm<0><<<grid, 256, 0, stream>>>(x, adj, W, coef, accum, out);
}